// SoftmaxApprox_46273977647903
// MI455X (gfx1250) — compile-verified
//
#include <hip/hip_runtime.h>

// CDNA5 / gfx1250. wave32 only.
typedef float v2f __attribute__((ext_vector_type(2)));
typedef float v8f __attribute__((ext_vector_type(8)));

// ---------------------------------------------------------------------------
// FHE sign-approximation polynomials (odd minimax composites f4, g4)
// p(x) = x * (c0 + x^2*(c1 + x^2*(c2 + x^2*(c3 + x^2*c4))))
// ---------------------------------------------------------------------------
__device__ __forceinline__ float comp_poly(float x, float c0, float c1,
                                           float c2, float c3, float c4) {
    float x2 = x * x;
    float t = __builtin_fmaf(x2, c4, c3);
    t = __builtin_fmaf(x2, t, c2);
    t = __builtin_fmaf(x2, t, c1);
    t = __builtin_fmaf(x2, t, c0);
    return x * t;
}

__device__ __forceinline__ float sign_poly(float x) {
    // g4 o g4 o f4 o f4
    float g = comp_poly(x, 5.712890625f, -34.154296875f, 94.7412109375f,
                        -110.83203125f, 45.5302734375f);
    g = comp_poly(g, 5.712890625f, -34.154296875f, 94.7412109375f,
                  -110.83203125f, 45.5302734375f);
    float f = comp_poly(g, 2.4609375f, -3.28125f, 2.953125f,
                        -1.40625f, 0.2734375f);
    return comp_poly(f, 2.4609375f, -3.28125f, 2.953125f,
                     -1.40625f, 0.2734375f);
}

__device__ __forceinline__ float poly_max(float a, float b) {
    float d = a - b;
    return ((a + b) + d * sign_poly(d)) * 0.5f;
}

// ---------------------------------------------------------------------------
// Rotate-and-polymax pass: dst[j] = polymax(src[j], src[(j - shift) mod n])
// SCALE: first pass fuses the *0.01 input scaling.
// VECROT: shift % 4 == 0 -> rotated stream is float4-aligned.
// Working set (48 MiB) is L2-resident on MI455X (192 MB L2), so these 22
// passes run at L2 bandwidth, not HBM.
// ---------------------------------------------------------------------------
template <bool SCALE, bool VECROT>
__global__ void __launch_bounds__(256)
maxpass_kernel(const float* __restrict__ src, float* __restrict__ dst,
               int shift, int n) {
    const int mask = n - 1;
    int j = (blockIdx.x * 256 + threadIdx.x) * 4;

    float4 a = *(const float4*)(src + j);
    float4 r;
    if (VECROT) {
        int rj = (j - shift) & mask;  // power-of-2 n, 4-aligned since shift%4==0
        r = *(const float4*)(src + rj);
    } else {
        r.x = src[(j + 0 - shift) & mask];
        r.y = src[(j + 1 - shift) & mask];
        r.z = src[(j + 2 - shift) & mask];
        r.w = src[(j + 3 - shift) & mask];
    }
    if (SCALE) {
        a.x *= 0.01f; a.y *= 0.01f; a.z *= 0.01f; a.w *= 0.01f;
        r.x *= 0.01f; r.y *= 0.01f; r.z *= 0.01f; r.w *= 0.01f;
    }
    float4 o;
    o.x = poly_max(a.x, r.x);
    o.y = poly_max(a.y, r.y);
    o.z = poly_max(a.z, r.z);
    o.w = poly_max(a.w, r.w);
    *(float4*)(dst + j) = o;
}

// ---------------------------------------------------------------------------
// a_exp = (1 + (x - max)/2^10)^(2^10); also produce per-block partial sums.
// Wave-level reduction uses V_WMMA_F32_16X16X4_F32 with A = ones:
// each lane contributes 2 accumulators as B-matrix K-entries, so one matrix
// op collapses 4 values per column (lanes n and n+16, both VGPRs) into 16
// exact fp32 column sums; 4 shfl_xor adds finish the wave.
// ---------------------------------------------------------------------------
__global__ void __launch_bounds__(256)
exp_partial_kernel(const float* __restrict__ in, const float* __restrict__ amax,
                   float* __restrict__ aexp, float* __restrict__ partials) {
    __shared__ float wsum[8];
    const int tid = threadIdx.x;
    const int base = (blockIdx.x * 256 + tid) * 8;

    float4 x0 = *(const float4*)(in + base);
    float4 x1 = *(const float4*)(in + base + 4);
    float4 m0 = *(const float4*)(amax + base);
    float4 m1 = *(const float4*)(amax + base + 4);

    float e[8];
    float xs[8] = {x0.x, x0.y, x0.z, x0.w, x1.x, x1.y, x1.z, x1.w};
    float ms[8] = {m0.x, m0.y, m0.z, m0.w, m1.x, m1.y, m1.z, m1.w};
#pragma unroll
    for (int i = 0; i < 8; ++i) {
        // a_max was computed on 0.01-scaled data: rescale by *100
        float y = __builtin_fmaf(xs[i] - ms[i] * 100.0f, 1.0f / 1024.0f, 1.0f);
#pragma unroll
        for (int s = 0; s < 10; ++s) y = y * y;  // (1+x/1024)^1024
        e[i] = y;
    }
    float4 o0 = {e[0], e[1], e[2], e[3]};
    float4 o1 = {e[4], e[5], e[6], e[7]};
    *(float4*)(aexp + base) = o0;
    *(float4*)(aexp + base + 4) = o1;

    float acc0 = (e[0] + e[1]) + (e[2] + e[3]);
    float acc1 = (e[4] + e[5]) + (e[6] + e[7]);

    // --- WMMA reduction (EXEC all-1s here: no divergence above) ---
    v2f amat; amat.x = 1.0f; amat.y = 1.0f;       // A = ones (16x4)
    v2f bmat; bmat.x = acc0; bmat.y = acc1;       // B = data  (4x16)
    v8f cacc = {};
    cacc = __builtin_amdgcn_wmma_f32_16x16x4_f32(
        /*neg_a=*/false, amat, /*neg_b=*/false, bmat,
        /*c_mod=*/(short)0, cacc, /*reuse_a=*/false, /*reuse_b=*/false);
    // lane L now holds colsum(L & 15) in cacc[0] (replicated across halves)
    float s = cacc[0];
    s += __shfl_xor(s, 1, 32);
    s += __shfl_xor(s, 2, 32);
    s += __shfl_xor(s, 4, 32);
    s += __shfl_xor(s, 8, 32);   // every lane: sum of this wave's 256 elems

    const int wid = tid >> 5;
    if ((tid & 31) == 0) wsum[wid] = s;
    __syncthreads();
    if (tid == 0) {
        float t = 0.0f;
#pragma unroll
        for (int w = 0; w < 8; ++w) t += wsum[w];  // fixed order: deterministic
        partials[blockIdx.x] = t;
    }
}

// ---------------------------------------------------------------------------
// Deterministic final reduction + Goldschmidt inverse of sum_exp*(2/dim).
// ---------------------------------------------------------------------------
__global__ void __launch_bounds__(256)
reduce_invert_kernel(const float* __restrict__ partials, float* __restrict__ inv_out,
                     int p, float two_over_dim) {
    __shared__ float sm[256];
    const int tid = threadIdx.x;
    float s = 0.0f;
    for (int i = tid; i < p; i += 256) s += partials[i];
    sm[tid] = s;
    __syncthreads();
    for (int off = 128; off > 0; off >>= 1) {
        if (tid < off) sm[tid] += sm[tid + off];
        __syncthreads();
    }
    if (tid == 0) {
        float x = sm[0] * two_over_dim;          // sum_exp slot value
        float a = 2.0f - x;
        float b = 1.0f - x;
#pragma unroll
        for (int i = 0; i < 5; ++i) {            // Goldschmidt, d = 5
            b = b * b;
            a = a * (b + 1.0f);
        }
        inv_out[0] = a * two_over_dim;           // invert(sum_exp)*(2/dim)
    }
}

__global__ void __launch_bounds__(256)
scale_kernel(float* __restrict__ out, const float* __restrict__ inv, int n) {
    (void)n;
    const float iv = inv[0];
    int j = (blockIdx.x * 256 + threadIdx.x) * 4;
    float4 v = *(const float4*)(out + j);
    v.x *= iv; v.y *= iv; v.z *= iv; v.w *= iv;
    *(float4*)(out + j) = v;
}

// ---------------------------------------------------------------------------
// Launch: 22 polymax passes ping-ponging d_out <-> ws, then exp+partials,
// scalar invert, broadcast scale. All on `stream`, graph-capture safe.
// ---------------------------------------------------------------------------
extern "C" void kernel_launch(void* const* d_in, const int* in_sizes, int n_in,
                              void* d_out, int out_size, void* d_ws, size_t ws_size,
                              hipStream_t stream) {
    (void)n_in; (void)out_size; (void)ws_size;
    const float* in = (const float*)d_in[0];
    const int n = in_sizes[0];                 // 1<<22, power of two; dim == n
    float* out = (float*)d_out;
    float* ws0 = (float*)d_ws;                 // n floats: max-phase ping-pong
    const int pblocks = n / (256 * 8);         // exp blocks / #partials
    float* partials = ws0 + n;
    float* inv = partials + pblocks;

    const int log2n = 31 - __builtin_clz((unsigned)n);
    const int blocks4 = n / (4 * 256);

    // MAX phase: rotate-tree of polynomial maxes (L2-resident streams).
    // pass k: shift = 2^k; dst = (k odd) ? ws0 : out
    maxpass_kernel<true, false><<<blocks4, 256, 0, stream>>>(in, out, 1, n);
    maxpass_kernel<false, false><<<blocks4, 256, 0, stream>>>(out, ws0, 2, n);
    for (int k = 2; k < log2n; ++k) {
        const float* s = ((k - 1) & 1) ? ws0 : out;
        float* d = (k & 1) ? ws0 : out;
        maxpass_kernel<false, true><<<blocks4, 256, 0, stream>>>(s, d, 1 << k, n);
    }
    const float* amax = ((log2n - 1) & 1) ? ws0 : out;   // final pass dest

    // EXP + per-block partial sums (WMMA-assisted wave reduction)
    exp_partial_kernel<<<pblocks, 256, 0, stream>>>(in, amax, out, partials);

    // Scalar total sum + Goldschmidt inverse
    const float two_over_dim = 2.0f / (float)n;
    reduce_invert_kernel<<<1, 256, 0, stream>>>(partials, inv, pblocks, two_over_dim);

    // out = a_exp * inverted
    scale_kernel<<<blocks4, 256, 0, stream>>>(out, inv, n);
}